// NetGCN_89215060673014
// MI455X (gfx1250) — compile-verified
//
#include <hip/hip_runtime.h>
#include <hip/hip_bf16.h>

// ---------------------------------------------------------------------------
// GCN (3x GCNConv + BN + ReLU, final linear) for MI455X / gfx1250, wave32.
//   N=50000 nodes, E=1.6M edges, feature dims 128, edge dim 16, fp32.
//
// Roofline: edge gather (h[row]: 819MB) + scatter atomics (819MB) + edge_attr
// (102MB) per layer ~= 1.8GB x3 ~= 5.4GB -> ~0.23ms @ 23.3 TB/s. Compute is
// ~25 GFLOP total => memory bound. Design minimizes HBM traffic:
//   * edge features computed on-the-fly from LDS-resident We (8x saving)
//   * BN+ReLU folded algebraically into the next GEMM's weights:
//       (relu(x)*a + b) @ W == relu(x) @ (a_k*W[k][n]) + (bias[n] + sum_k b_k*W[k][n])
//     so the WMMA inner loop is pure load->relu->wmma (no affine loads)
//   * node buffers (25.6MB) ping-pong in workspace, L2-resident (192MB L2)
// Node GEMMs use V_WMMA_F32_16X16X4_F32 (fp32-precise), full 128x128 W in
// 64KB LDS; compiler hoists all 32 B-fragments into VGPRs (verified in asm).
// ---------------------------------------------------------------------------

typedef __attribute__((ext_vector_type(2))) float v2f;
typedef __attribute__((ext_vector_type(8))) float v8f;

#define HID 128
#define DEDGE 16
#define BN_EPS 1e-5f

// ---------------------------------------------------------------------------
// fill: generic value fill (acc zeroing, deg=1 init, stats zeroing)
// ---------------------------------------------------------------------------
__global__ void fill_kernel(float* __restrict__ p, float v, long n) {
    long i = (long)blockIdx.x * blockDim.x + threadIdx.x;
    long stride = (long)gridDim.x * blockDim.x;
    for (; i < n; i += stride) p[i] = v;
}

// deg[row[e]] += 1  (deg pre-filled with 1.0 for the +1.0 in reference)
__global__ void deg_count_kernel(const int* __restrict__ row, float* __restrict__ deg, int E) {
    int i = blockIdx.x * blockDim.x + threadIdx.x;
    int stride = gridDim.x * blockDim.x;
    for (; i < E; i += stride) atomicAdd(&deg[row[i]], 1.0f);
}

// deg -> deg^-0.5 in place
__global__ void dinv_kernel(float* __restrict__ deg, int n) {
    int i = blockIdx.x * blockDim.x + threadIdx.x;
    if (i < n) deg[i] = rsqrtf(deg[i]);
}

// ---------------------------------------------------------------------------
// WMMA fp32 GEMM:  C[M,128] = op(A)[M,128] @ W[128,128] + bias
//   RELU=true applies relu to A elements (BN affine already folded into W/bias)
// Block = 256 threads = 8 waves; wave w owns N-tile [16w,16w+16).
// Full W staged in 64KB LDS once per block; block strides over M tiles.
// wave32 16x16 tile layout: A frag v2f (lanes 0-15: K={0,1}; 16-31: K={2,3}),
// B frag v2f symmetric, C/D v8f (VGPR r: M=r for lanes 0-15, M=r+8 for 16-31).
// ---------------------------------------------------------------------------
template <bool RELU>
__global__ void gemm_wmma_kernel(const float* __restrict__ A,
                                 const float* __restrict__ W,
                                 const float* __restrict__ bias,
                                 float* __restrict__ C, int M) {
    __shared__ float sW[HID * HID];   // 64 KB
    int tid = threadIdx.x;
    for (int i = tid; i < HID * HID; i += 256) sW[i] = W[i];
    __syncthreads();

    const int lane  = tid & 31;
    const int wave  = tid >> 5;
    const int n0    = wave * 16;
    const int mloc  = lane & 15;
    const int khalf = (lane >> 4) * 2;          // 0 or 2
    const int nn    = n0 + (lane & 15);
    const float bv  = bias[nn];
    const int mtiles = M >> 4;

    for (int mt = blockIdx.x; mt < mtiles; mt += gridDim.x) {
        const int m0 = mt * 16;
        const float* __restrict__ arow = A + (long)(m0 + mloc) * HID + khalf;
        v8f acc = {0.f, 0.f, 0.f, 0.f, 0.f, 0.f, 0.f, 0.f};

        #pragma unroll
        for (int k4 = 0; k4 < HID / 4; ++k4) {
            const int k = k4 * 4 + khalf;
            float a0 = arow[k4 * 4];
            float a1 = arow[k4 * 4 + 1];
            if (RELU) {
                a0 = fmaxf(a0, 0.f);
                a1 = fmaxf(a1, 0.f);
            }
            v2f av; av.x = a0; av.y = a1;
            v2f bw; bw.x = sW[k * HID + nn]; bw.y = sW[(k + 1) * HID + nn];
            acc = __builtin_amdgcn_wmma_f32_16x16x4_f32(
                /*neg_a=*/false, av, /*neg_b=*/false, bw,
                /*c_mod=*/(short)0, acc, /*reuse_a=*/false, /*reuse_b=*/false);
        }

        const int mbase = m0 + ((lane >> 4) << 3);  // +8 for upper half-lanes
        #pragma unroll
        for (int r = 0; r < 8; ++r)
            C[(long)(mbase + r) * HID + nn] = acc[r] + bv;
    }
}

// ---------------------------------------------------------------------------
// Edge kernel: one wave per edge, lane handles 4 channels (j=4*lane..+3).
//   f  = edge_attr[e] @ We + be        (We 16x128 in 8KB LDS, shfl-broadcast)
//   msg= dinv[row]*dinv[col] * relu(h[row] + f)
//   atomicAdd into acc[col]
// ---------------------------------------------------------------------------
__global__ void edge_kernel(const float* __restrict__ h,
                            const float* __restrict__ ea,
                            const int* __restrict__ row,
                            const int* __restrict__ col,
                            const float* __restrict__ We,
                            const float* __restrict__ be,
                            const float* __restrict__ dinv,
                            float* __restrict__ acc, int E) {
    __shared__ float sWe[DEDGE * HID];  // 8 KB
    __shared__ float sbe[HID];
    int tid = threadIdx.x;
    for (int i = tid; i < DEDGE * HID; i += 256) sWe[i] = We[i];
    if (tid < HID) sbe[tid] = be[tid];
    __syncthreads();

    const int lane = tid & 31;
    const int wave = tid >> 5;
    const int j = lane * 4;

    for (long e = (long)blockIdx.x * 8 + wave; e < E; e += (long)gridDim.x * 8) {
        const int r = row[e], c = col[e];
        const float nrm = dinv[r] * dinv[c];
        const float eav = ea[e * DEDGE + (lane & 15)];

        float f0 = sbe[j], f1 = sbe[j + 1], f2 = sbe[j + 2], f3 = sbe[j + 3];
        #pragma unroll
        for (int k = 0; k < DEDGE; ++k) {
            const float ek = __shfl(eav, k, 32);
            const float* wrow = &sWe[k * HID + j];
            f0 = fmaf(ek, wrow[0], f0);
            f1 = fmaf(ek, wrow[1], f1);
            f2 = fmaf(ek, wrow[2], f2);
            f3 = fmaf(ek, wrow[3], f3);
        }
        const float4 hv = *(const float4*)(h + (long)r * HID + j);
        float* dst = acc + (long)c * HID + j;
        atomicAdd(dst + 0, nrm * fmaxf(hv.x + f0, 0.f));
        atomicAdd(dst + 1, nrm * fmaxf(hv.y + f1, 0.f));
        atomicAdd(dst + 2, nrm * fmaxf(hv.z + f2, 0.f));
        atomicAdd(dst + 3, nrm * fmaxf(hv.w + f3, 0.f));
    }
}

// ---------------------------------------------------------------------------
// BN stats over relu(acc): per-channel sum -> s[0:128), sumsq -> s[128:256)
// ---------------------------------------------------------------------------
__global__ void bn_stats_kernel(const float* __restrict__ acc,
                                float* __restrict__ s, int n) {
    __shared__ float red1[256];
    __shared__ float red2[256];
    const int tid = threadIdx.x;
    const int ch = tid & (HID - 1);
    const int sub = tid >> 7;  // 0 or 1 (two rows in flight per block)
    float s1 = 0.f, s2 = 0.f;
    for (long i = (long)blockIdx.x * 2 + sub; i < n; i += (long)gridDim.x * 2) {
        const float v = fmaxf(acc[i * HID + ch], 0.f);
        s1 += v;
        s2 += v * v;
    }
    red1[tid] = s1;
    red2[tid] = s2;
    __syncthreads();
    if (sub == 0) {
        atomicAdd(&s[ch],        s1 + red1[tid + 128]);
        atomicAdd(&s[HID + ch],  s2 + red2[tid + 128]);
    }
}

// stats -> affine:  a = gamma*rsqrt(var+eps) at s[256:384), b = beta-mu*a at s[384:512)
__global__ void bn_finalize_kernel(float* __restrict__ s,
                                   const float* __restrict__ gamma,
                                   const float* __restrict__ beta, float invN) {
    const int ch = threadIdx.x;
    const float mu  = s[ch] * invN;
    const float var = s[HID + ch] * invN - mu * mu;
    const float a   = rsqrtf(var + BN_EPS) * gamma[ch];
    s[2 * HID + ch] = a;
    s[3 * HID + ch] = beta[ch] - mu * a;
}

// ---------------------------------------------------------------------------
// Fold BN affine (a,b) of the previous layer into the next layer's weights:
//   Wp[k][n] = a[k] * W[k][n]
//   cp[n]    = bias[n] + sum_k b[k] * W[k][n]
// so  (relu(x)*a + b) @ W + bias == relu(x) @ Wp + cp   (exact algebra).
// One block, 128 threads (thread = output column n); 16K FMAs, negligible.
// ---------------------------------------------------------------------------
__global__ void fold_kernel(const float* __restrict__ W,
                            const float* __restrict__ bias,
                            const float* __restrict__ aff,   // a[0:128), b[128:256)
                            float* __restrict__ Wp,
                            float* __restrict__ cp) {
    const int n = threadIdx.x;
    float c = bias[n];
    #pragma unroll 4
    for (int k = 0; k < HID; ++k) {
        const float w = W[k * HID + n];
        Wp[k * HID + n] = aff[k] * w;
        c = fmaf(aff[HID + k], w, c);
    }
    cp[n] = c;
}

// ---------------------------------------------------------------------------
// Host-side orchestration
// ---------------------------------------------------------------------------
extern "C" void kernel_launch(void* const* d_in, const int* in_sizes, int n_in,
                              void* d_out, int out_size, void* d_ws, size_t ws_size,
                              hipStream_t stream) {
    const float* x    = (const float*)d_in[0];
    const float* ea   = (const float*)d_in[1];
    const int*   eidx = (const int*)d_in[2];
    const float* W1  = (const float*)d_in[3];
    const float* b1  = (const float*)d_in[4];
    const float* We1 = (const float*)d_in[5];
    const float* be1 = (const float*)d_in[6];
    const float* g1  = (const float*)d_in[7];
    const float* bt1 = (const float*)d_in[8];
    const float* W2  = (const float*)d_in[9];
    const float* b2  = (const float*)d_in[10];
    const float* We2 = (const float*)d_in[11];
    const float* be2 = (const float*)d_in[12];
    const float* g2  = (const float*)d_in[13];
    const float* bt2 = (const float*)d_in[14];
    const float* W3  = (const float*)d_in[15];
    const float* b3  = (const float*)d_in[16];
    const float* We3 = (const float*)d_in[17];
    const float* be3 = (const float*)d_in[18];
    const float* g3  = (const float*)d_in[19];
    const float* bt3 = (const float*)d_in[20];
    const float* Wl  = (const float*)d_in[21];
    const float* bl  = (const float*)d_in[22];

    const int N = in_sizes[0] / HID;     // 50000
    const int E = in_sizes[1] / DEDGE;   // 1600000
    const int* row = eidx;
    const int* col = eidx + E;

    // workspace layout (floats)
    float* ws   = (float*)d_ws;
    float* bufA = ws;                        // N*128  (h = GEMM output)
    float* bufB = bufA + (long)N * HID;      // N*128  (aggregation accumulator)
    float* dinv = bufB + (long)N * HID;      // N
    float* stat = dinv + N;                  // 512: s1|s2|a|b
    float* Wp   = stat + 4 * HID;            // 128*128 folded weights
    float* cp   = Wp + HID * HID;            // 128 folded bias

    const long NH = (long)N * HID;
    const float invN = 1.0f / (float)N;
    const int gemmGrid  = 1024;
    const int edgeGrid  = (E + 7) / 8;
    const int statGrid  = 2048;
    const int fillGrid  = 4096;

    // degree -> dinv (shared by all 3 conv layers)
    fill_kernel<<<256, 256, 0, stream>>>(dinv, 1.0f, (long)N);
    deg_count_kernel<<<2048, 256, 0, stream>>>(row, dinv, E);
    dinv_kernel<<<(N + 255) / 256, 256, 0, stream>>>(dinv, N);

    // ---- layer 1 ----
    gemm_wmma_kernel<false><<<gemmGrid, 256, 0, stream>>>(x, W1, b1, bufA, N);
    fill_kernel<<<fillGrid, 256, 0, stream>>>(bufB, 0.0f, NH);
    edge_kernel<<<edgeGrid, 256, 0, stream>>>(bufA, ea, row, col, We1, be1, dinv, bufB, E);
    fill_kernel<<<1, 256, 0, stream>>>(stat, 0.0f, 2 * HID);
    bn_stats_kernel<<<statGrid, 256, 0, stream>>>(bufB, stat, N);
    bn_finalize_kernel<<<1, HID, 0, stream>>>(stat, g1, bt1, invN);

    // ---- layer 2 ----  (BN1+ReLU folded into W2')
    fold_kernel<<<1, HID, 0, stream>>>(W2, b2, stat + 2 * HID, Wp, cp);
    gemm_wmma_kernel<true><<<gemmGrid, 256, 0, stream>>>(bufB, Wp, cp, bufA, N);
    fill_kernel<<<fillGrid, 256, 0, stream>>>(bufB, 0.0f, NH);
    edge_kernel<<<edgeGrid, 256, 0, stream>>>(bufA, ea, row, col, We2, be2, dinv, bufB, E);
    fill_kernel<<<1, 256, 0, stream>>>(stat, 0.0f, 2 * HID);
    bn_stats_kernel<<<statGrid, 256, 0, stream>>>(bufB, stat, N);
    bn_finalize_kernel<<<1, HID, 0, stream>>>(stat, g2, bt2, invN);

    // ---- layer 3 ----  (BN2+ReLU folded into W3')
    fold_kernel<<<1, HID, 0, stream>>>(W3, b3, stat + 2 * HID, Wp, cp);
    gemm_wmma_kernel<true><<<gemmGrid, 256, 0, stream>>>(bufB, Wp, cp, bufA, N);
    fill_kernel<<<fillGrid, 256, 0, stream>>>(bufB, 0.0f, NH);
    edge_kernel<<<edgeGrid, 256, 0, stream>>>(bufA, ea, row, col, We3, be3, dinv, bufB, E);
    fill_kernel<<<1, 256, 0, stream>>>(stat, 0.0f, 2 * HID);
    bn_stats_kernel<<<statGrid, 256, 0, stream>>>(bufB, stat, N);
    bn_finalize_kernel<<<1, HID, 0, stream>>>(stat, g3, bt3, invN);

    // ---- final linear ----  (BN3+ReLU folded into Wl')
    fold_kernel<<<1, HID, 0, stream>>>(Wl, bl, stat + 2 * HID, Wp, cp);
    gemm_wmma_kernel<true><<<gemmGrid, 256, 0, stream>>>(bufB, Wp, cp, (float*)d_out, N);
}